// VectorQuantizer_25099788878510
// MI455X (gfx1250) — compile-verified
//
#include <hip/hip_runtime.h>
#include <hip/hip_bf16.h>

typedef __attribute__((ext_vector_type(2))) float v2f;
typedef __attribute__((ext_vector_type(8))) float v8f;

#define NUM_EMB       512
#define EMB_DIM       64
#define N_ROWS        131072      // 32*64*64 spatial positions
#define CH_STRIDE     4096        // 64*64 (stride between channels, NCHW)
#define BATCH_STRIDE  262144      // 64*64*64
#define ROWS_PER_BLK  128
#define QUANT_ELEMS   8388608     // 32*64*64*64
#define WB_STRIDE     68          // padded LDS row stride (floats): bank = (4j+c)%64, conflict-free

// ---------------------------------------------------------------------------
// Kernel 0: codebook squared norms + zero the cross-block accumulators.
// ---------------------------------------------------------------------------
__global__ void vq_prep(const float* __restrict__ w, float* __restrict__ wnorm,
                        int* __restrict__ counts, double* __restrict__ sqsum) {
  int j = blockIdx.x * blockDim.x + threadIdx.x;
  if (j < NUM_EMB) {
    float s = 0.f;
    #pragma unroll 8
    for (int c = 0; c < EMB_DIM; ++c) { float v = w[j * EMB_DIM + c]; s += v * v; }
    wnorm[j]  = s;
    counts[j] = 0;
  }
  if (j == 0) *sqsum = 0.0;
}

// ---------------------------------------------------------------------------
// Kernel 1: distances via V_WMMA_F32_16X16X4_F32, argmin, gather, loss, hist.
// 256 threads = 8 waves; each wave owns a 16-row A tile; block owns 128 rows.
// Loss uses sum_c (q-x)^2 = ||x_row||^2 + best_d  (best_d = ||w||^2 - 2 x.w),
// so x is read exactly once from HBM.
// ---------------------------------------------------------------------------
__global__ void __launch_bounds__(256)
vq_main(const float* __restrict__ x, const float* __restrict__ w,
        const float* __restrict__ wnorm, int* __restrict__ counts,
        double* __restrict__ sqsum, float* __restrict__ out_q,
        float* __restrict__ out_idx) {
  __shared__ float wbuf[128 * WB_STRIDE];   // 128 codebook rows, padded
  __shared__ int   idx_lds[ROWS_PER_BLK];
  __shared__ float bestd_lds[ROWS_PER_BLK];
  __shared__ float xnorm_lds[ROWS_PER_BLK];
  __shared__ int   hcount[NUM_EMB];
  __shared__ float red[256];

  const int tid   = threadIdx.x;
  const int lane  = tid & 31;
  const int wv    = tid >> 5;       // wave id 0..7
  const int khalf = lane >> 4;      // 0: K even pair, 1: K odd pair (A layout)
  const int lrow  = lane & 15;

  hcount[tid]       = 0;
  hcount[tid + 256] = 0;

  // ---- A tile: rows n0..n0+15, all 64 channels, in WMMA f32 16x16x4 layout.
  // x[n, c] lives at NCHW addr (n>>12)*BATCH_STRIDE + c*CH_STRIDE + (n&4095).
  // Simultaneously accumulate this lane's half of ||x_row||^2.
  const int n0    = blockIdx.x * ROWS_PER_BLK + wv * 16;
  const int n     = n0 + lrow;
  const int abase = (n >> 12) * BATCH_STRIDE + (n & 4095);
  v2f areg[16];
  float xn_part = 0.f;
  #pragma unroll
  for (int kk = 0; kk < 16; ++kk) {
    const int c0 = 4 * kk + 2 * khalf;
    const float ax = x[abase + c0 * CH_STRIDE];
    const float ay = x[abase + (c0 + 1) * CH_STRIDE];
    areg[kk].x = ax;
    areg[kk].y = ay;
    xn_part += ax * ax + ay * ay;
  }
  // other half of the row's channels lives in lane^16
  const float xn = xn_part + __shfl_xor(xn_part, 16, 32);
  if (khalf == 0) xnorm_lds[wv * 16 + lrow] = xn;

  float bestv[8]; int besti[8];
  #pragma unroll
  for (int r = 0; r < 8; ++r) { bestv[r] = 3.4e38f; besti[r] = 0; }

  for (int chunk = 0; chunk < 4; ++chunk) {
    __syncthreads();                       // previous chunk's compute done
    #pragma unroll
    for (int i = 0; i < 32; ++i) {         // stage 128x64 codebook rows
      const int e = tid + i * 256;
      const int jl = e >> 6, c = e & 63;
      wbuf[jl * WB_STRIDE + c] = w[(chunk * 128 + jl) * EMB_DIM + c];
    }
    __syncthreads();

    #pragma unroll
    for (int tn = 0; tn < 8; ++tn) {       // 8 N-tiles of 16 codes
      const int jl = tn * 16 + lrow;
      const int jg = chunk * 128 + jl;
      v8f acc = {};
      #pragma unroll
      for (int kk = 0; kk < 16; ++kk) {    // K = 64 in steps of 4
        const int c0 = 4 * kk + 2 * khalf;
        v2f b;
        b.x = wbuf[jl * WB_STRIDE + c0];
        b.y = wbuf[jl * WB_STRIDE + c0 + 1];
        acc = __builtin_amdgcn_wmma_f32_16x16x4_f32(
            false, areg[kk], false, b, (short)0, acc, false, false);
      }
      const float wn = wnorm[jg];          // ||x||^2 constant per row: dropped
      #pragma unroll
      for (int r = 0; r < 8; ++r) {
        const float d = wn - 2.0f * acc[r];
        if (d < bestv[r]) { bestv[r] = d; besti[r] = jg; }
      }
    }
  }

  // ---- argmin across the 16 lanes of each half-wave (C layout: lane = col)
  #pragma unroll
  for (int r = 0; r < 8; ++r) {
    float bv = bestv[r]; int bi = besti[r];
    #pragma unroll
    for (int m = 8; m >= 1; m >>= 1) {
      const float ov = __shfl_xor(bv, m, 32);
      const int   oi = __shfl_xor(bi, m, 32);
      if (ov < bv || (ov == bv && oi < bi)) { bv = ov; bi = oi; }
    }
    bestv[r] = bv; besti[r] = bi;
  }
  if (lrow == 0) {                         // lane 0: rows r; lane 16: rows r+8
    #pragma unroll
    for (int r = 0; r < 8; ++r) {
      const int rl = wv * 16 + khalf * 8 + r;
      idx_lds[rl]   = besti[r];
      bestd_lds[rl] = bestv[r];
      out_idx[blockIdx.x * ROWS_PER_BLK + rl] = (float)besti[r];
    }
  }
  __syncthreads();

  // ---- block-local histogram, flushed with global atomics
  if (tid < ROWS_PER_BLK) atomicAdd(&hcount[idx_lds[tid]], 1);
  __syncthreads();
  #pragma unroll
  for (int i = 0; i < 2; ++i) {
    const int bin = tid + i * 256;
    const int cnt = hcount[bin];
    if (cnt) atomicAdd(&counts[bin], cnt);
  }

  // ---- gather codebook rows -> quantized output (NCHW); no x re-read
  #pragma unroll 4
  for (int i = 0; i < 32; ++i) {
    const int e  = tid + i * 256;          // 8192 = 64 channels * 128 rows
    const int c  = e >> 7, rl = e & 127;
    const int nn = blockIdx.x * ROWS_PER_BLK + rl;
    const int addr = (nn >> 12) * BATCH_STRIDE + c * CH_STRIDE + (nn & 4095);
    out_q[addr] = w[idx_lds[rl] * EMB_DIM + c];
  }

  // ---- loss partial: per-row ||x||^2 + best_d, reduced across the block
  red[tid] = (tid < ROWS_PER_BLK) ? (xnorm_lds[tid] + bestd_lds[tid]) : 0.f;
  __syncthreads();
  for (int s = 128; s > 0; s >>= 1) {
    if (tid < s) red[tid] += red[tid + s];
    __syncthreads();
  }
  if (tid == 0) atomicAdd(sqsum, (double)red[0]);
}

// ---------------------------------------------------------------------------
// Kernel 2: loss scalar + perplexity from counts.
// ---------------------------------------------------------------------------
__global__ void vq_finalize(const int* __restrict__ counts,
                            const double* __restrict__ sqsum,
                            float* __restrict__ out) {
  __shared__ float red[256];
  const int tid = threadIdx.x;
  float local = 0.f;
  for (int i = tid; i < NUM_EMB; i += 256) {
    const float p = (float)counts[i] * (1.0f / (float)N_ROWS);
    local += p * logf(p + 1e-10f);
  }
  red[tid] = local;
  __syncthreads();
  for (int s = 128; s > 0; s >>= 1) {
    if (tid < s) red[tid] += red[tid + s];
    __syncthreads();
  }
  if (tid == 0) {
    out[0] = 1.25f * (float)(*sqsum * (1.0 / (double)QUANT_ELEMS));  // q + 0.25*e
    out[QUANT_ELEMS + 1] = expf(-red[0]);                            // perplexity
  }
}

// ---------------------------------------------------------------------------
extern "C" void kernel_launch(void* const* d_in, const int* in_sizes, int n_in,
                              void* d_out, int out_size, void* d_ws, size_t ws_size,
                              hipStream_t stream) {
  const float* x = (const float*)d_in[0];   // [32,64,64,64] NCHW
  const float* w = (const float*)d_in[1];   // [512,64]
  float* out = (float*)d_out;               // [loss | quant | ppl | indices]
  char* ws = (char*)d_ws;
  double* sqsum = (double*)ws;                    // 8 B
  float*  wnorm = (float*)(ws + 64);              // 512 f32
  int*    counts = (int*)(ws + 64 + 2048);        // 512 i32

  vq_prep<<<2, 256, 0, stream>>>(w, wnorm, counts, sqsum);
  vq_main<<<N_ROWS / ROWS_PER_BLK, 256, 0, stream>>>(
      x, w, wnorm, counts, sqsum,
      /*out_q=*/out + 1, /*out_idx=*/out + QUANT_ELEMS + 2);
  vq_finalize<<<1, 256, 0, stream>>>(counts, sqsum, out);
}